// TorchDecoding_79542794322704
// MI455X (gfx1250) — compile-verified
//
#include <hip/hip_runtime.h>

// ---------------------------------------------------------------------------
// Beam-search backtrace (gather_tree) for MI455X / gfx1250.
//
// shapes: step_ids / parent_ids : [max_len, batch, beam=8] int32
//         seq_lens              : [batch, beam] int32
// out:    ids  : [batch, beam, max_len] int32 (contiguous per (b,k) thread)
//         lens : [batch, beam] int32
//
// One thread per (batch, beam). Dependency chain cur <- parent[t,b,cur] is
// broken by loading the full 8-beam group (32B, addr independent of cur) and
// selecting in registers (3-level cndmask tree). 4x time unroll -> 16
// independent b128 loads in flight per chunk; output written as int4 stores
// into the thread's private contiguous row.
// ---------------------------------------------------------------------------

__device__ __forceinline__ int sel8(int4 a, int4 b, int idx) {
    const bool s0 = (idx & 1) != 0;
    const bool s1 = (idx & 2) != 0;
    const bool s2 = (idx & 4) != 0;
    int r01 = s0 ? a.y : a.x;
    int r23 = s0 ? a.w : a.z;
    int r45 = s0 ? b.y : b.x;
    int r67 = s0 ? b.w : b.z;
    int rl  = s1 ? r23 : r01;
    int rh  = s1 ? r67 : r45;
    return s2 ? rh : rl;
}

__global__ __launch_bounds__(256) void gather_tree_kernel(
    const int* __restrict__ step_ids,
    const int* __restrict__ parent_ids,
    const int* __restrict__ seq_lens,
    const int* __restrict__ end_id_ptr,
    int* __restrict__ ids_out,
    int* __restrict__ len_out,
    int batch, int max_len)
{
    const int tid = blockIdx.x * blockDim.x + threadIdx.x;
    const int b   = tid >> 3;
    const int k   = tid & 7;
    if (b >= batch) return;

    const int end_id = *end_id_ptr;          // uniform scalar load
    const int bk     = batch * 8;            // elements per time-step

    // msl = min(max_k seq_lens[b,k], max_len)
    const int4* sl = reinterpret_cast<const int4*>(seq_lens + (size_t)b * 8);
    int4 l0 = sl[0];
    int4 l1 = sl[1];
    int m = l0.x;
    m = l0.y > m ? l0.y : m;  m = l0.z > m ? l0.z : m;  m = l0.w > m ? l0.w : m;
    m = l1.x > m ? l1.x : m;  m = l1.y > m ? l1.y : m;
    m = l1.z > m ? l1.z : m;  m = l1.w > m ? l1.w : m;
    const int msl = m < max_len ? m : max_len;

    int* row = ids_out + (size_t)tid * (size_t)max_len;   // private contiguous row
    const size_t gb = (size_t)b * 8;

    const int4 endv = make_int4(end_id, end_id, end_id, end_id);

    // chunk base of the topmost (possibly partial) chunk touched by backtrace
    const int c0 = (msl > 0) ? ((msl - 1) & ~3) : 0;

    // Pre-fill the all-end tail [c0+4, max_len) with int4 stores.
    int tt = (msl > 0) ? (c0 + 4) : 0;
    for (; tt + 4 <= max_len; tt += 4)
        *reinterpret_cast<int4*>(row + tt) = endv;
    for (; tt < max_len; ++tt) row[tt] = end_id;

    int cur       = k;      // start on own beam
    int first_end = msl;    // positions t >= msl are end tokens already

    if (msl > 0) {
        // --- peel: top partial chunk [c0, c0+3] ---
        for (int t = c0 + 3; t >= msl; --t) row[t] = end_id;
        for (int t = msl - 1; t >= c0; --t) {
            const size_t g = (size_t)t * bk + gb;
            const int4* sg = reinterpret_cast<const int4*>(step_ids   + g);
            const int4* pg = reinterpret_cast<const int4*>(parent_ids + g);
            int4 sa = sg[0], sb = sg[1];
            int4 pa = pg[0], pb = pg[1];
            const int tok = sel8(sa, sb, cur);
            cur = sel8(pa, pb, cur);
            if (tok == end_id) first_end = t;
            row[t] = tok;
        }

        // --- main: full chunks of 4 time-steps, 16 independent b128 loads ---
        for (int t4 = c0 - 4; t4 >= 0; t4 -= 4) {
            const size_t g0 = (size_t)t4 * bk + gb;
            const size_t g1 = g0 + bk;
            const size_t g2 = g1 + bk;
            const size_t g3 = g2 + bk;

            // gfx1250 global_prefetch_b8: pull t4-16 lines toward the WGP.
            if (t4 >= 16) {
                const size_t gp = (size_t)(t4 - 16) * bk + gb;
                __builtin_prefetch(step_ids   + gp, 0, 1);
                __builtin_prefetch(parent_ids + gp, 0, 1);
            }

            const int4* s3p = reinterpret_cast<const int4*>(step_ids   + g3);
            const int4* p3p = reinterpret_cast<const int4*>(parent_ids + g3);
            const int4* s2p = reinterpret_cast<const int4*>(step_ids   + g2);
            const int4* p2p = reinterpret_cast<const int4*>(parent_ids + g2);
            const int4* s1p = reinterpret_cast<const int4*>(step_ids   + g1);
            const int4* p1p = reinterpret_cast<const int4*>(parent_ids + g1);
            const int4* s0p = reinterpret_cast<const int4*>(step_ids   + g0);
            const int4* p0p = reinterpret_cast<const int4*>(parent_ids + g0);

            int4 sa3 = s3p[0], sb3 = s3p[1], pa3 = p3p[0], pb3 = p3p[1];
            int4 sa2 = s2p[0], sb2 = s2p[1], pa2 = p2p[0], pb2 = p2p[1];
            int4 sa1 = s1p[0], sb1 = s1p[1], pa1 = p1p[0], pb1 = p1p[1];
            int4 sa0 = s0p[0], sb0 = s0p[1], pa0 = p0p[0], pb0 = p0p[1];

            const int tok3 = sel8(sa3, sb3, cur);  cur = sel8(pa3, pb3, cur);
            const int tok2 = sel8(sa2, sb2, cur);  cur = sel8(pa2, pb2, cur);
            const int tok1 = sel8(sa1, sb1, cur);  cur = sel8(pa1, pb1, cur);
            const int tok0 = sel8(sa0, sb0, cur);  cur = sel8(pa0, pb0, cur);

            if (tok3 == end_id) first_end = t4 + 3;
            if (tok2 == end_id) first_end = t4 + 2;
            if (tok1 == end_id) first_end = t4 + 1;
            if (tok0 == end_id) first_end = t4;

            *reinterpret_cast<int4*>(row + t4) = make_int4(tok0, tok1, tok2, tok3);
        }
    }

    // Truncate after first end token (usually an empty loop: random tokens
    // rarely equal end_id, and [msl, max_len) is already end-filled).
    for (int t = first_end; t < msl; ++t) row[t] = end_id;

    len_out[tid] = first_end;
}

extern "C" void kernel_launch(void* const* d_in, const int* in_sizes, int n_in,
                              void* d_out, int out_size, void* d_ws, size_t ws_size,
                              hipStream_t stream) {
    (void)n_in; (void)out_size; (void)d_ws; (void)ws_size;

    const int* output_ids = (const int*)d_in[0];
    const int* parent_ids = (const int*)d_in[1];
    const int* seq_lens   = (const int*)d_in[2];
    const int* end_id     = (const int*)d_in[3];

    const int bk      = in_sizes[2];        // batch * beam (beam == 8)
    const int batch   = bk / 8;
    const int max_len = in_sizes[0] / bk;   // 256

    int* ids_out = (int*)d_out;
    int* len_out = ids_out + (size_t)bk * (size_t)max_len;

    const int threads = bk;                 // one thread per (batch, beam)
    const int block   = 256;                // 8 wave32s per workgroup
    const int grid    = (threads + block - 1) / block;

    gather_tree_kernel<<<grid, block, 0, stream>>>(
        output_ids, parent_ids, seq_lens, end_id,
        ids_out, len_out, batch, max_len);
}